// EmbeddingGATEdgeV2_24575802867854
// MI455X (gfx1250) — compile-verified
//
#include <hip/hip_runtime.h>
#include <hip/hip_bf16.h>
#include <math.h>

#define N_NODES   50000
#define N_EDGES   1200000
#define EMBED_DIM 64
#define SLOPE     0.2f
#define LN_EPS    1e-5f

// ---- d_out layout (float slots). Edge-output region doubles as big scratch
// until the final kernel overwrites it.
#define EB        3200000                  // edge output region start (floats)
#define S_SIZE    (N_NODES * 400)          // S[node][head][nf_j] : 20,000,000
#define S_OFF     (EB)
#define EIJ_OFF   (EB + 20000000)          // eij[E][4] : 4,800,000

// ---- d_ws layout (32-bit words), total ~264K words (~1.06 MB)
#define WC_OFF    0                        // WC[100][128] = [W1 | W2]
#define ELN_OFF   12800                    // edge LN table [12][64]
#define EE_OFF    13568                    // edge attn part [12][4] (attn_b folded)
#define SMAX_OFF  13616                    // ordered-uint seg max [N][4]
#define DEG_OFF   213616                   // int deg[N]

typedef float v2f __attribute__((ext_vector_type(2)));
typedef float v8f __attribute__((ext_vector_type(8)));

__device__ __forceinline__ unsigned f2ord(float f) {
    unsigned u = __float_as_uint(f);
    return (u & 0x80000000u) ? ~u : (u | 0x80000000u);
}
__device__ __forceinline__ float ord2f(unsigned u) {
    return (u & 0x80000000u) ? __uint_as_float(u & 0x7fffffffu) : __uint_as_float(~u);
}
__device__ __forceinline__ float lrelu(float x) { return x > 0.f ? x : SLOPE * x; }

// ---------------- K0a: zero the S accumulator (in d_out edge region) --------
__global__ void k_zero_s(float4* __restrict__ s4, int n4) {
    int i = blockIdx.x * blockDim.x + threadIdx.x;
    if (i < n4) s4[i] = make_float4(0.f, 0.f, 0.f, 0.f);
}

// ---------------- K0b: zero seg-max keys and degree counters ----------------
__global__ void k_zero_ws(unsigned* __restrict__ smax, int* __restrict__ deg) {
    int i = blockIdx.x * blockDim.x + threadIdx.x;
    if (i < N_NODES * 4) smax[i] = 0u;           // ordered key 0 < key(-inf)
    if (i < N_NODES)     deg[i]  = 0;
}

// ---------------- K1: WC = node_emb @ [w_W_top | w_W_bot]  via f32 WMMA -----
// C[112,128] tiles: block m = row tile (7), wave n = col tile (8), K = 64.
__global__ __launch_bounds__(256) void k_wc_wmma(const float* __restrict__ node_emb,
                                                 const float* __restrict__ w_W,
                                                 float* __restrict__ WC) {
    int m    = blockIdx.x;          // 0..6
    int n    = threadIdx.x >> 5;    // 0..7 (wave id)
    int lane = threadIdx.x & 31;
    int hi   = lane >> 4;           // 0 or 1 (half-wave)
    int l15  = lane & 15;
    int arow = m * 16 + l15;        // A: M index lives in lanes
    int col  = n * 16 + l15;        // B/C: N index lives in lanes
    v8f acc = {0.f, 0.f, 0.f, 0.f, 0.f, 0.f, 0.f, 0.f};
    for (int k0 = 0; k0 < 64; k0 += 4) {
        int kh = k0 + hi * 2;       // K split: low lanes K+0/1, high lanes K+2/3
        v2f a, b;
        if (arow < 100) { a.x = node_emb[arow * 64 + kh]; a.y = node_emb[arow * 64 + kh + 1]; }
        else            { a.x = 0.f;                      a.y = 0.f; }
        // Bcat[k][c] : c<64 -> w_W[k][c] (W1 half) ; c>=64 -> w_W[64+k][c-64] (W2 half)
        b.x = (col < 64) ? w_W[kh * 64 + col]       : w_W[(64 + kh) * 64 + (col - 64)];
        b.y = (col < 64) ? w_W[(kh + 1) * 64 + col] : w_W[(65 + kh) * 64 + (col - 64)];
        acc = __builtin_amdgcn_wmma_f32_16x16x4_f32(false, a, false, b, (short)0, acc,
                                                    false, false);
    }
#pragma unroll
    for (int v = 0; v < 8; v++) {
        int row = m * 16 + v + hi * 8;   // C/D: VGPR v -> M=v (low lanes) / v+8 (high)
        if (row < 100) WC[row * 128 + col] = acc[v];
    }
}

// ---------------- K2: 12-row edge tables: edge_h -> LN row + attn partial ---
__global__ __launch_bounds__(64) void k_edge_tables(
    const float* __restrict__ edge_emb, const float* __restrict__ edgew_W,
    const float* __restrict__ edgew_b, const float* __restrict__ attn_W,
    const float* __restrict__ attn_b, const float* __restrict__ ln_g,
    const float* __restrict__ ln_b, float* __restrict__ ELN, float* __restrict__ EE) {
    __shared__ float row[64], r1[64], r2[64];
    int d = blockIdx.x, f = threadIdx.x;
    float acc = edgew_b[f];
    for (int k = 0; k < 64; k++) acc += edge_emb[d * 64 + k] * edgew_W[k * 64 + f];
    row[f] = acc; r1[f] = acc; r2[f] = acc * acc;
    __syncthreads();
    for (int off = 32; off > 0; off >>= 1) {
        if (f < off) { r1[f] += r1[f + off]; r2[f] += r2[f + off]; }
        __syncthreads();
    }
    float mu  = r1[0] * (1.f / 64.f);
    float var = r2[0] * (1.f / 64.f) - mu * mu;
    ELN[d * 64 + f] = (acc - mu) / sqrtf(var + LN_EPS) * ln_g[f] + ln_b[f];
    if (f < 4) {
        float s = attn_b[0];
        for (int k = 0; k < 16; k++) s += lrelu(row[f * 16 + k]) * attn_W[16 + k];
        EE[d * 4 + f] = s;
    }
}

// ---------------- K3: per-edge eij + seg-max + degree -----------------------
__global__ __launch_bounds__(256) void k_edge_eij(
    const int* __restrict__ nfeat, const int* __restrict__ efeat,
    const int* __restrict__ eidx, const float* __restrict__ WC,
    const float* __restrict__ w_b, const float* __restrict__ attn_W,
    const float* __restrict__ EE, float* __restrict__ EIJ,
    unsigned* __restrict__ smax, int* __restrict__ deg) {
    __shared__ float sWC[100 * 132];   // stride 132 to spread LDS banks
    __shared__ float sB[64];
    __shared__ float sA[16];
    __shared__ float sE[48];
    int tid = threadIdx.x;
    for (int i = tid; i < 12800; i += 256) {
        int r = i >> 7, c = i & 127;
        sWC[r * 132 + c] = WC[i];
    }
    if (tid < 64) sB[tid] = w_b[tid];
    if (tid < 16) sA[tid] = attn_W[tid];
    if (tid < 48) sE[tid] = EE[tid];
    __syncthreads();
    int e = blockIdx.x * 256 + tid;
    if (e >= N_EDGES) return;
    int s = eidx[e], t = eidx[N_EDGES + e];
    int ni = nfeat[t], nj = nfeat[s], ee = efeat[e];
    const float* r1 = &sWC[ni * 132];        // W1 row (h_i part)
    const float* r2 = &sWC[nj * 132 + 64];   // W2 row (h_j part)
    float res[4];
#pragma unroll
    for (int h = 0; h < 4; h++) {
        float acc = sE[ee * 4 + h];
#pragma unroll
        for (int k = 0; k < 16; k++) {
            int f = h * 16 + k;
            acc += lrelu(r1[f] + r2[f] + sB[f]) * sA[k];
        }
        res[h] = acc;
        atomicMax(&smax[t * 4 + h], f2ord(acc));
    }
    ((float4*)EIJ)[e] = make_float4(res[0], res[1], res[2], res[3]);
    atomicAdd(&deg[t], 1);
}

// ---------------- K4: ex = exp(eij - max) scattered into S[t][h][nf_j] ------
__global__ __launch_bounds__(256) void k_scatter(
    const int* __restrict__ nfeat, const int* __restrict__ eidx,
    const float* __restrict__ EIJ, const unsigned* __restrict__ smax,
    float* __restrict__ S) {
    int e = blockIdx.x * 256 + threadIdx.x;
    if (e >= N_EDGES) return;
    int s = eidx[e], t = eidx[N_EDGES + e];
    int nj = nfeat[s];
    float4 ev = ((const float4*)EIJ)[e];
    float evs[4] = {ev.x, ev.y, ev.z, ev.w};
    float* Sn = S + (size_t)t * 400 + nj;
#pragma unroll
    for (int h = 0; h < 4; h++) {
        float m = ord2f(smax[t * 4 + h]);
        atomicAdd(&Sn[h * 100], __expf(evs[h] - m));
    }
}

// ---------------- K5: per-node combine (S @ emb)/sum + deg*emb, LayerNorm ---
__global__ __launch_bounds__(256) void k_node_out(
    const int* __restrict__ nfeat, const float* __restrict__ node_emb,
    const float* __restrict__ S, const int* __restrict__ deg,
    const float* __restrict__ ln_g, const float* __restrict__ ln_b,
    float* __restrict__ out) {
    __shared__ float sEmb[6400];
    __shared__ float sS[1600];
    __shared__ float red1[256], red2[256];
    int tid = threadIdx.x;
    for (int i = tid; i < 6400; i += 256) sEmb[i] = node_emb[i];
    const float* Sb = S + (size_t)blockIdx.x * 1600;
    for (int i = tid; i < 1600; i += 256) sS[i] = Sb[i];
    __syncthreads();
    int ln = tid >> 6, f = tid & 63, h = f >> 4;
    int n = blockIdx.x * 4 + ln;   // 50000 % 4 == 0 -> always valid
    float ssum = 0.f, acc = 0.f;
    const float* Srow = &sS[ln * 400 + h * 100];
#pragma unroll 4
    for (int d = 0; d < 100; d++) {
        float sv = Srow[d];
        ssum += sv;
        acc += sv * sEmb[d * 64 + f];
    }
    int dg = deg[n];
    int nn = nfeat[n];
    float o = (dg > 0) ? (acc / ssum + (float)dg * sEmb[nn * 64 + f]) : 0.f;
    red1[tid] = o; red2[tid] = o * o;
    __syncthreads();
    for (int off = 32; off > 0; off >>= 1) {
        if (f < off) { red1[tid] += red1[tid + off]; red2[tid] += red2[tid + off]; }
        __syncthreads();
    }
    float mu  = red1[ln * 64] * (1.f / 64.f);
    float var = red2[ln * 64] * (1.f / 64.f) - mu * mu;
    out[(size_t)n * 64 + f] = (o - mu) / sqrtf(var + LN_EPS) * ln_g[f] + ln_b[f];
}

// ---------------- K6: edge output = broadcast of 12 LN'd rows ---------------
__global__ __launch_bounds__(256) void k_edge_out(const int* __restrict__ efeat,
                                                  const float* __restrict__ ELN,
                                                  float4* __restrict__ out4) {
    __shared__ float4 sT[192];   // 12 rows x 16 float4
    int tid = threadIdx.x;
    if (tid < 192) sT[tid] = ((const float4*)ELN)[tid];
    __syncthreads();
    long long i = (long long)blockIdx.x * 256 + tid;   // float4 index
    if (i >= (long long)N_EDGES * 16) return;
    int e = (int)(i >> 4), q = (int)(i & 15);
    out4[i] = sT[efeat[e] * 16 + q];
}

extern "C" void kernel_launch(void* const* d_in, const int* in_sizes, int n_in,
                              void* d_out, int out_size, void* d_ws, size_t ws_size,
                              hipStream_t stream) {
    const int*   node_features = (const int*)d_in[0];
    const int*   edge_features = (const int*)d_in[1];
    const int*   edge_index    = (const int*)d_in[2];
    const float* node_emb      = (const float*)d_in[3];
    const float* edge_emb      = (const float*)d_in[4];
    const float* w_W           = (const float*)d_in[5];
    const float* w_b           = (const float*)d_in[6];
    const float* edgew_W       = (const float*)d_in[7];
    const float* edgew_b       = (const float*)d_in[8];
    const float* attn_W        = (const float*)d_in[9];
    const float* attn_b        = (const float*)d_in[10];
    const float* ln_g          = (const float*)d_in[11];
    const float* ln_b          = (const float*)d_in[12];

    float* out = (float*)d_out;
    float* ws  = (float*)d_ws;

    float*    WC   = ws + WC_OFF;
    float*    ELN  = ws + ELN_OFF;
    float*    EE   = ws + EE_OFF;
    unsigned* SMAX = (unsigned*)(ws + SMAX_OFF);
    int*      DEG  = (int*)(ws + DEG_OFF);
    float*    S    = out + S_OFF;
    float*    EIJ  = out + EIJ_OFF;

    int n4 = S_SIZE / 4;   // 5,000,000 float4
    k_zero_s<<<(n4 + 255) / 256, 256, 0, stream>>>((float4*)S, n4);
    k_zero_ws<<<(N_NODES * 4 + 255) / 256, 256, 0, stream>>>(SMAX, DEG);
    k_wc_wmma<<<7, 256, 0, stream>>>(node_emb, w_W, WC);
    k_edge_tables<<<12, 64, 0, stream>>>(edge_emb, edgew_W, edgew_b, attn_W, attn_b,
                                         ln_g, ln_b, ELN, EE);
    k_edge_eij<<<(N_EDGES + 255) / 256, 256, 0, stream>>>(node_features, edge_features,
                                                          edge_index, WC, w_b, attn_W,
                                                          EE, EIJ, SMAX, DEG);
    k_scatter<<<(N_EDGES + 255) / 256, 256, 0, stream>>>(node_features, edge_index,
                                                         EIJ, SMAX, S);
    k_node_out<<<N_NODES / 4, 256, 0, stream>>>(node_features, node_emb, S, DEG,
                                                ln_g, ln_b, out);
    k_edge_out<<<(int)(((long long)N_EDGES * 16 + 255) / 256), 256, 0, stream>>>(
        edge_features, ELN, (float4*)(out + EB));
}